// LearnableLMWTLayer_51857435131907
// MI455X (gfx1250) — compile-verified
//
#include <hip/hip_runtime.h>
#include <math.h>

// ---------------------------------------------------------------------------
// Fused 4-level learnable wavelet transform (DWT -> gated details -> IWT)
// for x[B=4, S=8192, D=1024] f32 on gfx1250 (MI455X).
//
// Bandwidth-bound: 256 MB mandatory HBM traffic @ 23.3 TB/s ~= 11 us. The
// whole pyramid is computed in one LDS-resident pass per (b, 16ch, 512t)
// tile. TDM (tensor_load_to_lds / tensor_store_from_lds) does the strided
// [time x 16ch] tile DMA both ways; threads own 4 channels x 2 adjacent
// positions so all LDS traffic is b128 and conv taps are shared.
// ---------------------------------------------------------------------------

typedef unsigned int u32x4 __attribute__((ext_vector_type(4)));
typedef int          i32x8 __attribute__((ext_vector_type(8)));
typedef int          i32x4 __attribute__((ext_vector_type(4)));
typedef float        f32x4 __attribute__((ext_vector_type(4)));

#define TCHUNK 512      // output samples per workgroup
#define SLEN   8192
#define DCH    1024
#define XLEN   640      // TCHUNK + 2*64 halo

// LDS float offsets (layout [pos][16 channels], 64B rows)
#define OFF_X   0        // 640*16 = 10240  (reused as rec0 staging for TDM store)
#define OFF_A1  10240    // 316*16 = 5056
#define OFF_A2  15296    // 156*16 = 2496
#define OFF_A3  17792    //  76*16 = 1216
#define OFF_A4  19008    //  36*16 =  576
#define OFF_D0  19584
#define OFF_D1  24640
#define OFF_D2  27136
#define OFF_D3  28352
#define SMEM_FLOATS 28928   // 115712 bytes

#if __has_builtin(__builtin_amdgcn_tensor_load_to_lds) && \
    __has_builtin(__builtin_amdgcn_tensor_store_from_lds)
#define USE_TDM 1
#else
#define USE_TDM 0
#endif

__device__ __forceinline__ float sigm(float v) { return 1.0f / (1.0f + __expf(-v)); }

#if USE_TDM
// 2D tile <-> LDS DMA: tile = rows x 16 f32, global row stride = DCH floats.
__device__ __forceinline__ void tdm_tile_2d(bool is_load, unsigned ldsb,
                                            unsigned long long ga, int rows)
{
    u32x4 g0;
    g0[0] = 1u;                                           // count=1, user mode
    g0[1] = ldsb;                                         // lds_addr (bytes)
    g0[2] = (unsigned)(ga & 0xFFFFFFFFull);               // global_addr lo
    g0[3] = (unsigned)((ga >> 32) & 0x1FFFFFFull) | (2u << 30);  // hi | type=2

    i32x8 g1;
    g1[0] = (2 << 16);                                    // data_size = 4B
    g1[1] = (16 << 16);                                   // tensor_dim0 = 16 (lo16)
    g1[2] = (int)((unsigned)rows << 16);                  // tensor_dim1 lo16
    g1[3] = (int)(((unsigned)rows >> 16) | (16u << 16));  // dim1 hi | tile_dim0=16
    g1[4] = (rows & 0xFFFF);                              // tile_dim1=rows, tile_dim2=0
    g1[5] = DCH;                                          // tensor_dim0_stride
    g1[6] = 0;
    g1[7] = 0;

    i32x4 gz = {0, 0, 0, 0};
#if defined(__clang_major__) && __clang_major__ >= 23
    i32x8 g4 = {0, 0, 0, 0, 0, 0, 0, 0};
    if (is_load) __builtin_amdgcn_tensor_load_to_lds(g0, g1, gz, gz, g4, 0);
    else         __builtin_amdgcn_tensor_store_from_lds(g0, g1, gz, gz, g4, 0);
#else
    if (is_load) __builtin_amdgcn_tensor_load_to_lds(g0, g1, gz, gz, 0);
    else         __builtin_amdgcn_tensor_store_from_lds(g0, g1, gz, gz, 0);
#endif
}
#endif

__global__ void __launch_bounds__(256)
lmwt_fused_kernel(const float* __restrict__ x,
                  const float* __restrict__ lpf,    // [4,5]
                  const float* __restrict__ syf,    // [4,5]
                  const float* __restrict__ gw,     // [4,1024]
                  float* __restrict__ out)
{
    constexpr int H[5]    = {64, 30, 14, 6, 2};         // analysis halos
    constexpr int LEN[5]  = {XLEN, 316, 156, 76, 36};   // buffer lengths per level
    constexpr int LENR[4] = {512, 258, 132, 68};        // reconstruction extents
    constexpr int RL[4]   = {0, 1, 2, 2};               // rec halo per level
    constexpr int CL[4]   = {1, 2, 2, 2};               // A-side read base offset

    extern __shared__ float smem[];
    float* buf[5]  = { smem + OFF_X,  smem + OFF_A1, smem + OFF_A2,
                       smem + OFF_A3, smem + OFF_A4 };
    float* dbuf[4] = { smem + OFF_D0, smem + OFF_D1, smem + OFF_D2, smem + OFF_D3 };

    const int tid  = threadIdx.x;
    const int cg   = tid & 3;       // channel group of 4 (ch = 4*cg .. 4*cg+3)
    const int pos0 = tid >> 2;      // position slot (0..63)
    const int t0   = blockIdx.x * TCHUNK;
    const int d0   = blockIdx.y * 16;
    const int b    = blockIdx.z;
    const f32x4 z4 = {0.0f, 0.0f, 0.0f, 0.0f};

    // ---------------- global -> LDS tile (time x 16ch) --------------------
    const int tlo  = t0 - H[0];
    const int vlo  = (tlo < 0) ? 0 : tlo;
    int vhi        = tlo + XLEN; if (vhi > SLEN) vhi = SLEN;
    const int rows = vhi - vlo;

    if (rows != XLEN) {   // edge chunk: zero-fill off-tensor halo
        f32x4* xv = (f32x4*)(smem + OFF_X);
        for (int i = tid; i < XLEN * 4; i += 256) xv[i] = z4;
    }
    __syncthreads();

#if USE_TDM
    if ((tid >> 5) == 0) {   // one wave issues the DMA
        const float* gp = x + ((size_t)b * SLEN + (size_t)vlo) * DCH + d0;
        unsigned ldsb = (unsigned)(size_t)(smem + OFF_X + (size_t)(vlo - tlo) * 16);
        tdm_tile_2d(true, ldsb, (unsigned long long)(size_t)gp, rows);
        __builtin_amdgcn_s_wait_tensorcnt(0);
    }
#else
    for (int i = tid; i < rows * 4; i += 256) {
        const int r = i >> 2, c = (i & 3) * 4;
        *(f32x4*)(smem + OFF_X + (size_t)(vlo - tlo + r) * 16 + c) =
            *(const f32x4*)(x + ((size_t)b * SLEN + (size_t)(vlo + r)) * DCH + d0 + c);
    }
#endif
    __syncthreads();

    // ---------------- analysis: 4 levels of DWT + gated details -----------
#pragma unroll
    for (int l = 0; l < 4; ++l) {
        const float f0 = lpf[l * 5 + 0], f1 = lpf[l * 5 + 1], f2 = lpf[l * 5 + 2],
                    f3 = lpf[l * 5 + 3], f4 = lpf[l * 5 + 4];
        // QMF high-pass: h[k] = (-1)^k * f[4-k]
        const float h0 =  f4, h1 = -f3, h2 =  f2, h3 = -f1, h4 =  f0;
        const f32x4 gv = *(const f32x4*)(gw + l * DCH + d0 + 4 * cg);
        const f32x4 gate = { sigm(gv[0]), sigm(gv[1]), sigm(gv[2]), sigm(gv[3]) };

        const float* in = buf[l];
        float* oa = buf[l + 1];
        float* od = dbuf[l];
        const int n2    = LEN[l + 1] / 2;           // position pairs
        const int ioff  = (l == 0) ? 2 : 0;
        const int gbase = (t0 >> (l + 1)) - H[l + 1];
        const int lsz   = SLEN >> (l + 1);

        for (int q = pos0; q < n2; q += 64) {
            // outputs j=2q and j=2q+1 share taps: inputs rows 4q+ioff .. +6
            const float* r = in + (size_t)(4 * q + ioff) * 16 + 4 * cg;
            const f32x4 x0 = *(const f32x4*)(r);
            const f32x4 x1 = *(const f32x4*)(r + 16);
            const f32x4 x2 = *(const f32x4*)(r + 32);
            const f32x4 x3 = *(const f32x4*)(r + 48);
            const f32x4 x4 = *(const f32x4*)(r + 64);
            const f32x4 x5 = *(const f32x4*)(r + 80);
            const f32x4 x6 = *(const f32x4*)(r + 96);

            f32x4 caE = f0*x0 + f1*x1 + f2*x2 + f3*x3 + f4*x4;
            f32x4 cdE = h0*x0 + h1*x1 + h2*x2 + h3*x3 + h4*x4;
            f32x4 caO = f0*x2 + f1*x3 + f2*x4 + f3*x5 + f4*x6;
            f32x4 cdO = h0*x2 + h1*x3 + h2*x4 + h3*x5 + h4*x6;

            const int g = gbase + 2 * q;          // per-level zero padding masks
            if (g < 0 || g >= lsz)         { caE = z4; cdE = z4; }
            if (g + 1 < 0 || g + 1 >= lsz) { caO = z4; cdO = z4; }

            float* oap = oa + (size_t)(2 * q) * 16 + 4 * cg;
            *(f32x4*)(oap)      = caE;
            *(f32x4*)(oap + 16) = caO;
            float* odp = od + (size_t)(2 * q) * 16 + 4 * cg;
            *(f32x4*)(odp)      = cdE * gate;
            *(f32x4*)(odp + 16) = cdO * gate;
        }
        __syncthreads();
    }

    // ---------------- synthesis: 4 levels of IWT --------------------------
#pragma unroll
    for (int ll = 0; ll < 4; ++ll) {
        const int l = 3 - ll;
        const float s0 = syf[l * 5 + 0], s1 = syf[l * 5 + 1], s2 = syf[l * 5 + 2],
                    s3 = syf[l * 5 + 3], s4 = syf[l * 5 + 4];
        const float q0 =  s4, q1 = -s3, q2 =  s2, q3 = -s1, q4 =  s0;
        const float sf[5] = { s0, s1, s2, s3, s4 };
        const float qf[5] = { q0, q1, q2, q3, q4 };

        const float* ra = buf[l + 1];   // approx / previous reconstruction
        const float* rd = dbuf[l];      // gated details
        float* ro = buf[l];             // rec output (l>0); buf[0] = TDM staging
        const int n2 = LENR[l] / 2;
        const int R  = RL[l];
        const int C  = CL[l];
        const int HD = H[l + 1];
        const int gbase = (t0 >> l) - R;
        const int lsz   = SLEN >> l;

        for (int qq = pos0; qq < n2; qq += 64) {
            f32x4 accE = z4, accO = z4;   // outputs j=2qq (even) and 2qq+1 (odd)
#pragma unroll
            for (int k = 0; k < 5; ++k) {
                // parity of active taps is compile-time per output after unroll
                const bool evenTap = (((k + R) & 1) == 0);
                const int  di = evenTap ? (k - 2 - R) / 2 : (k - 1 - R) / 2;
                const f32x4 av = *(const f32x4*)(ra + (size_t)(qq + di + C)  * 16 + 4 * cg);
                const f32x4 dv = *(const f32x4*)(rd + (size_t)(qq + di + HD) * 16 + 4 * cg);
                if (evenTap) accE += sf[k] * av + qf[k] * dv;
                else         accO += sf[k] * av + qf[k] * dv;
            }

            if (l == 0) {
#if USE_TDM
                float* op = ro + (size_t)(2 * qq) * 16 + 4 * cg;   // stage for TDM store
                *(f32x4*)(op)      = accE;
                *(f32x4*)(op + 16) = accO;
#else
                float* op = out + ((size_t)b * SLEN + (size_t)(t0 + 2 * qq)) * DCH + d0 + 4 * cg;
                *(f32x4*)(op)       = accE;
                *(f32x4*)(op + DCH) = accO;
#endif
            } else {
                const int g = gbase + 2 * qq;
                if (g < 0 || g >= lsz)         accE = z4;
                if (g + 1 < 0 || g + 1 >= lsz) accO = z4;
                float* op = ro + (size_t)(2 * qq) * 16 + 4 * cg;
                *(f32x4*)(op)      = accE;
                *(f32x4*)(op + 16) = accO;
            }
        }
        __syncthreads();
    }

#if USE_TDM
    // ---------------- LDS -> global tile via TDM --------------------------
    if ((tid >> 5) == 0) {
        const float* gp = out + ((size_t)b * SLEN + (size_t)t0) * DCH + d0;
        unsigned ldsb = (unsigned)(size_t)(smem + OFF_X);
        tdm_tile_2d(false, ldsb, (unsigned long long)(size_t)gp, TCHUNK);
        __builtin_amdgcn_s_wait_tensorcnt(0);   // drain before LDS dealloc/endpgm
    }
#endif
}

extern "C" void kernel_launch(void* const* d_in, const int* in_sizes, int n_in,
                              void* d_out, int out_size, void* d_ws, size_t ws_size,
                              hipStream_t stream) {
    const float* x   = (const float*)d_in[0];
    const float* lpf = (const float*)d_in[1];
    const float* syf = (const float*)d_in[2];
    const float* gw  = (const float*)d_in[3];
    float* out = (float*)d_out;

    const size_t shmem = (size_t)SMEM_FLOATS * sizeof(float);   // 113 KB < 320 KB
    (void)hipFuncSetAttribute((const void*)lmwt_fused_kernel,
                              hipFuncAttributeMaxDynamicSharedMemorySize,
                              (int)shmem);

    dim3 grid(SLEN / TCHUNK, DCH / 16, 4);   // 16 x 64 x 4 = 4096 workgroups
    dim3 block(256);
    lmwt_fused_kernel<<<grid, block, shmem, stream>>>(x, lpf, syf, gw, out);

    (void)in_sizes; (void)n_in; (void)out_size; (void)d_ws; (void)ws_size;
}